// GFocalV2Head_25855703122126
// MI455X (gfx1250) — compile-verified
//
#include <hip/hip_runtime.h>
#include <math.h>

typedef __attribute__((ext_vector_type(2))) float v2f;
typedef __attribute__((ext_vector_type(8))) float v8f;

#define MTOT 131072      // 32 * 64 * 64 rows
#define KTOT 256
#define KC   64          // K chunk staged in LDS
#define LDA  68          // padded LDS row stride (floats)

// ---------------------------------------------------------------------------
// GEMM: Y[M, Nout] = X[M,256] @ W[Nout,256]^T + bias  (fp32, WMMA 16x16x4)
// Block: 256 threads = 8 waves. Block tile: 128 rows x 64 cols.
// Wave w owns rows [w*16, w*16+16), four 16x16 N-subtiles (4 indep acc chains).
// ---------------------------------------------------------------------------
__global__ __launch_bounds__(256) void gemm_wmma_f32(
    const float* __restrict__ X, const float* __restrict__ W,
    const float* __restrict__ bias, float* __restrict__ Y, int Nout)
{
  __shared__ float As[128 * LDA];   // 34816 B
  __shared__ float Bs[64 * LDA];    // 17408 B

  const int tid  = threadIdx.x;
  const int wave = tid >> 5;
  const int lane = tid & 31;
  const int m    = lane & 15;       // row/col within fragment
  const int hi   = lane >> 4;       // 0 or 1
  const int kk   = hi * 2;          // A/B: lanes 16-31 hold K=2,3

  const long rowBase = (long)blockIdx.x * 128;
  const int  n0g     = blockIdx.y * 64;

  v8f acc0 = {0.f,0.f,0.f,0.f,0.f,0.f,0.f,0.f};
  v8f acc1 = acc0, acc2 = acc0, acc3 = acc0;

  for (int k0 = 0; k0 < KTOT; k0 += KC) {
    // stage A chunk: 128 x 64 floats = 2048 float4, 8 per thread
#pragma unroll
    for (int i = 0; i < 8; ++i) {
      int fi  = tid + i * 256;          // float4 index
      int r   = fi >> 4;                // row in tile
      int c4  = (fi & 15) << 2;         // col (float)
      float4 v = *(const float4*)(X + (rowBase + r) * KTOT + k0 + c4);
      *(float4*)&As[r * LDA + c4] = v;
    }
    // stage B chunk: 64 x 64 floats = 1024 float4, 4 per thread (guard Nout)
#pragma unroll
    for (int i = 0; i < 4; ++i) {
      int fi  = tid + i * 256;
      int r   = fi >> 4;
      int c4  = (fi & 15) << 2;
      float4 v = make_float4(0.f, 0.f, 0.f, 0.f);
      if (n0g + r < Nout)
        v = *(const float4*)(W + (long)(n0g + r) * KTOT + k0 + c4);
      *(float4*)&Bs[r * LDA + c4] = v;
    }
    __syncthreads();

    const float* ap  = &As[(wave * 16 + m) * LDA + kk];
    const float* bp0 = &Bs[(m     ) * LDA + kk];
    const float* bp1 = &Bs[(m + 16) * LDA + kk];
    const float* bp2 = &Bs[(m + 32) * LDA + kk];
    const float* bp3 = &Bs[(m + 48) * LDA + kk];
#pragma unroll
    for (int ks = 0; ks < KC; ks += 4) {
      v2f a  = *(const v2f*)(ap  + ks);
      v2f b0 = *(const v2f*)(bp0 + ks);
      v2f b1 = *(const v2f*)(bp1 + ks);
      v2f b2 = *(const v2f*)(bp2 + ks);
      v2f b3 = *(const v2f*)(bp3 + ks);
      acc0 = __builtin_amdgcn_wmma_f32_16x16x4_f32(false, a, false, b0,
                                                   (short)0, acc0, false, false);
      acc1 = __builtin_amdgcn_wmma_f32_16x16x4_f32(false, a, false, b1,
                                                   (short)0, acc1, false, false);
      acc2 = __builtin_amdgcn_wmma_f32_16x16x4_f32(false, a, false, b2,
                                                   (short)0, acc2, false, false);
      acc3 = __builtin_amdgcn_wmma_f32_16x16x4_f32(false, a, false, b3,
                                                   (short)0, acc3, false, false);
    }
    __syncthreads();
  }

  // store D: row = vgpr + 8*hi, col = lane&15
  const long outRow = rowBase + wave * 16;
  const int c0 = n0g + m;
  const int c1 = c0 + 16;
  const int c2 = c0 + 32;
  const int c3 = c0 + 48;
#pragma unroll
  for (int v = 0; v < 8; ++v) {
    int r = v + 8 * hi;
    float* yr = Y + (outRow + r) * (long)Nout;
    if (c0 < Nout) yr[c0] = acc0[v] + bias[c0];
    if (c1 < Nout) yr[c1] = acc1[v] + bias[c1];
    if (c2 < Nout) yr[c2] = acc2[v] + bias[c2];
    if (c3 < Nout) yr[c3] = acc3[v] + bias[c3];
  }
}

// ---------------------------------------------------------------------------
// Depthwise 3x3 conv (pad 1) + bias + exact-erf GELU.
// Layout: (n, l=h*64+w, c) channel-fastest. Block = 256 threads (one per
// channel), grid = (H=64, N=32). Sliding 3-column window in registers;
// every tap is a coalesced 1KB line across the block.
// ---------------------------------------------------------------------------
__global__ __launch_bounds__(256) void dwconv_gelu(
    const float* __restrict__ in, const float* __restrict__ wgt,
    const float* __restrict__ bias, float* __restrict__ out)
{
  const int c = threadIdx.x;
  const int h = blockIdx.x;
  const int n = blockIdx.y;
  const long img = (long)n * 4096;

  float k[9];
#pragma unroll
  for (int i = 0; i < 9; ++i) k[i] = wgt[c * 9 + i];
  const float bv = bias[c];

  const bool hasT = (h > 0), hasB = (h < 63);
  const float* rowT = in + (img + (long)(h - 1) * 64) * 256 + c;
  const float* rowM = in + (img + (long)h       * 64) * 256 + c;
  const float* rowB = in + (img + (long)(h + 1) * 64) * 256 + c;
  float* orow = out + (img + (long)h * 64) * 256 + c;

  float tP = 0.f, mP = 0.f, bP = 0.f;
  float tC = hasT ? rowT[0] : 0.f;
  float mC = rowM[0];
  float bC = hasB ? rowB[0] : 0.f;

  for (int wc = 0; wc < 64; ++wc) {
    float tN = 0.f, mN = 0.f, bN = 0.f;
    if (wc < 63) {
      int off = (wc + 1) * 256;
      tN = hasT ? rowT[off] : 0.f;
      mN = rowM[off];
      bN = hasB ? rowB[off] : 0.f;
    }
    float s = k[0]*tP + k[1]*tC + k[2]*tN
            + k[3]*mP + k[4]*mC + k[5]*mN
            + k[6]*bP + k[7]*bC + k[8]*bN + bv;
    // exact GELU: 0.5*x*(1+erf(x/sqrt(2)))
    float g = 0.5f * s * (1.0f + erff(s * 0.70710678118654752f));
    orow[wc * 256] = g;
    tP = tC; mP = mC; bP = bC;
    tC = tN; mC = mN; bC = bN;
  }
}

// ---------------------------------------------------------------------------
// fc3 for cls branch: y[m] = X[m,:] . w + b  (Nout = 1). One wave per row.
// ---------------------------------------------------------------------------
__global__ __launch_bounds__(256) void fc_dot256(
    const float* __restrict__ X, const float* __restrict__ w,
    const float* __restrict__ b, float* __restrict__ y)
{
  const int tid  = threadIdx.x;
  const int lane = tid & 31;
  const int wv   = tid >> 5;
  const long row = (long)blockIdx.x * 8 + wv;
  const float* xr = X + row * 256;
  float s = 0.f;
#pragma unroll
  for (int i = 0; i < 8; ++i) s += xr[lane + i * 32] * w[lane + i * 32];
#pragma unroll
  for (int off = 16; off > 0; off >>= 1) s += __shfl_down(s, off, 32);
  if (lane == 0) y[row] = s + b[0];
}

// ---------------------------------------------------------------------------
// Per-pixel post-processing: softmax(17) per side, top-4 + mean stats,
// rc1(20->64) relu, rc2(64->1) sigmoid quality, cls = sigmoid(logit)*q,
// boxes = DFL expectation / 16.
// ---------------------------------------------------------------------------
__global__ __launch_bounds__(256) void postproc(
    const float* __restrict__ regp,   // [131072][68]
    const float* __restrict__ clsl,   // [131072]
    const float* __restrict__ rc1w,   // [64][20]
    const float* __restrict__ rc1b,   // [64]
    const float* __restrict__ rc2w,   // [64]
    const float* __restrict__ rc2b,   // [1]
    float* __restrict__ cls_out,      // [131072]
    float* __restrict__ box_out)      // [131072][4]
{
  __shared__ float s_w1[1280];
  __shared__ float s_b1[64];
  __shared__ float s_w2[64];
  const int tid = threadIdx.x;
  for (int i = tid; i < 1280; i += 256) s_w1[i] = rc1w[i];
  if (tid < 64) { s_b1[tid] = rc1b[tid]; s_w2[tid] = rc2w[tid]; }
  __syncthreads();

  const long pix = (long)blockIdx.x * 256 + tid;
  const float* rp = regp + pix * 68;

  float stat[20];
  float box[4];
#pragma unroll
  for (int s = 0; s < 4; ++s) {
    float p[17];
    float mx = -1e30f;
#pragma unroll
    for (int j = 0; j < 17; ++j) { p[j] = rp[s * 17 + j]; mx = fmaxf(mx, p[j]); }
    float sum = 0.f;
#pragma unroll
    for (int j = 0; j < 17; ++j) { p[j] = expf(p[j] - mx); sum += p[j]; }
    float inv = 1.0f / sum;
    float t0 = -1.f, t1 = -1.f, t2 = -1.f, t3 = -1.f, dist = 0.f;
#pragma unroll
    for (int j = 0; j < 17; ++j) {
      float v = p[j] * inv;
      dist += v * (float)j;
      if      (v > t0) { t3 = t2; t2 = t1; t1 = t0; t0 = v; }
      else if (v > t1) { t3 = t2; t2 = t1; t1 = v; }
      else if (v > t2) { t3 = t2; t2 = v; }
      else if (v > t3) { t3 = v; }
    }
    stat[s * 5 + 0] = t0; stat[s * 5 + 1] = t1;
    stat[s * 5 + 2] = t2; stat[s * 5 + 3] = t3;
    stat[s * 5 + 4] = 0.25f * (t0 + t1 + t2 + t3);
    box[s] = dist * (1.0f / 16.0f);
  }

  float qs = rc2b[0];
  for (int o = 0; o < 64; ++o) {
    float a = s_b1[o];
#pragma unroll
    for (int c = 0; c < 20; ++c) a += stat[c] * s_w1[o * 20 + c];
    a = fmaxf(a, 0.f);
    qs += a * s_w2[o];
  }
  float q  = 1.0f / (1.0f + expf(-qs));
  float cl = (1.0f / (1.0f + expf(-clsl[pix]))) * q;

  cls_out[pix] = cl;
  box_out[pix * 4 + 0] = box[0];
  box_out[pix * 4 + 1] = box[1];
  box_out[pix * 4 + 2] = box[2];
  box_out[pix * 4 + 3] = box[3];
}

// ---------------------------------------------------------------------------
// Launch: reg branch -> reg_pred (written straight into d_out), cls branch,
// then fused post-processing. Ping-pong activation buffers in workspace.
// ---------------------------------------------------------------------------
extern "C" void kernel_launch(void* const* d_in, const int* in_sizes, int n_in,
                              void* d_out, int out_size, void* d_ws, size_t ws_size,
                              hipStream_t stream)
{
  const float* x = (const float*)d_in[0];
  // c branch: 1..10, r branch: 11..20
  const float* c_fc1_w = (const float*)d_in[1];
  const float* c_fc1_b = (const float*)d_in[2];
  const float* c_dw1_w = (const float*)d_in[3];
  const float* c_dw1_b = (const float*)d_in[4];
  const float* c_fc2_w = (const float*)d_in[5];
  const float* c_fc2_b = (const float*)d_in[6];
  const float* c_dw2_w = (const float*)d_in[7];
  const float* c_dw2_b = (const float*)d_in[8];
  const float* c_fc3_w = (const float*)d_in[9];
  const float* c_fc3_b = (const float*)d_in[10];
  const float* r_fc1_w = (const float*)d_in[11];
  const float* r_fc1_b = (const float*)d_in[12];
  const float* r_dw1_w = (const float*)d_in[13];
  const float* r_dw1_b = (const float*)d_in[14];
  const float* r_fc2_w = (const float*)d_in[15];
  const float* r_fc2_b = (const float*)d_in[16];
  const float* r_dw2_w = (const float*)d_in[17];
  const float* r_dw2_b = (const float*)d_in[18];
  const float* r_fc3_w = (const float*)d_in[19];
  const float* r_fc3_b = (const float*)d_in[20];
  const float* rc1_w   = (const float*)d_in[21];
  const float* rc1_b   = (const float*)d_in[22];
  const float* rc2_w   = (const float*)d_in[23];
  const float* rc2_b   = (const float*)d_in[24];

  const size_t ACT = (size_t)MTOT * 256;   // 33,554,432 floats = 128 MiB
  float* bufA = (float*)d_ws;
  float* bufB = bufA + ACT;
  float* clsl = bufB + ACT;                // 131072 floats

  float* out    = (float*)d_out;
  float* clsOut = out;                     // 131072
  float* boxOut = out + 131072;            // 524288
  float* regOut = out + 131072 + 524288;   // 131072 * 68

  const dim3 blk(256);
  const dim3 gemmGrid(MTOT / 128, 256 / 64);   // 1024 x 4
  const dim3 gemm68Grid(MTOT / 128, 2);        // 68 cols -> 2 tiles of 64
  const dim3 dwGrid(64, 32);

  // ---- reg branch: fc3 output IS the reg_pred output tensor ----
  gemm_wmma_f32<<<gemmGrid, blk, 0, stream>>>(x,    r_fc1_w, r_fc1_b, bufA, 256);
  dwconv_gelu <<<dwGrid,   blk, 0, stream>>>(bufA, r_dw1_w, r_dw1_b, bufB);
  gemm_wmma_f32<<<gemmGrid, blk, 0, stream>>>(bufB, r_fc2_w, r_fc2_b, bufA, 256);
  dwconv_gelu <<<dwGrid,   blk, 0, stream>>>(bufA, r_dw2_w, r_dw2_b, bufB);
  gemm_wmma_f32<<<gemm68Grid, blk, 0, stream>>>(bufB, r_fc3_w, r_fc3_b, regOut, 68);

  // ---- cls branch ----
  gemm_wmma_f32<<<gemmGrid, blk, 0, stream>>>(x,    c_fc1_w, c_fc1_b, bufA, 256);
  dwconv_gelu <<<dwGrid,   blk, 0, stream>>>(bufA, c_dw1_w, c_dw1_b, bufB);
  gemm_wmma_f32<<<gemmGrid, blk, 0, stream>>>(bufB, c_fc2_w, c_fc2_b, bufA, 256);
  dwconv_gelu <<<dwGrid,   blk, 0, stream>>>(bufA, c_dw2_w, c_dw2_b, bufB);
  fc_dot256   <<<MTOT / 8, blk, 0, stream>>>(bufB, c_fc3_w, c_fc3_b, clsl);

  // ---- fused post-processing ----
  postproc<<<MTOT / 256, blk, 0, stream>>>(regOut, clsl, rc1_w, rc1_b,
                                           rc2_w, rc2_b, clsOut, boxOut);
}